// NnqlmCnnBasedRNN_73770358276977
// MI455X (gfx1250) — compile-verified
//
#include <hip/hip_runtime.h>
#include <hip/hip_bf16.h>
#include <math.h>

#define NT 2          // tensors: q, a
#define BSZ 16
#define LSZ 64
#define DSZ 128
#define COLG 8        // column groups per (tensor,b)
#define COLS 16       // columns per group
#define NTHREADS 256
#define RPT 8         // rows per thread (128 rows / 16 thread-groups)
#define STRIDE 17     // padded LDS row stride (floats) to avoid bank conflicts
#define EPSV 1e-4f

typedef __attribute__((ext_vector_type(2))) float v2f;
typedef __attribute__((ext_vector_type(8))) float v8f;

// Hardware transcendental tanh on gfx1250 (V_TANH_F32); fallback to libm.
#if __has_builtin(__builtin_amdgcn_tanhf)
#define TANH(x) __builtin_amdgcn_tanhf(x)
#else
#define TANH(x) tanhf(x)
#endif

// gfx1250 async global->LDS copy (GLOBAL_LOAD_ASYNC_TO_LDS_B32, ASYNCcnt).
#if __has_builtin(__builtin_amdgcn_global_load_async_to_lds_b32) && \
    __has_builtin(__builtin_amdgcn_s_wait_asynccnt)
#define HAVE_ASYNC_LDS 1
#endif

#ifdef HAVE_ASYNC_LDS
typedef __attribute__((address_space(1))) int glb_int;
typedef __attribute__((address_space(3))) int lds_int;
__device__ __forceinline__ void async_copy_f32(float* lds_dst, const float* gsrc) {
    __builtin_amdgcn_global_load_async_to_lds_b32(
        (glb_int*)(uintptr_t)gsrc,
        (lds_int*)(unsigned)(uintptr_t)lds_dst,
        0, 0);
}
#endif

// ---------------------------------------------------------------------------
// Fused: embedding gather -> density (folded to rank-1 form) -> 2-layer
// conv-RNN scan -> max-pool over time. One block owns (tensor, b, 16 columns);
// columns are independent under the row-mixing recurrence, so no cross-block
// sync is needed. Hidden states live in LDS; max accumulators in VGPRs.
// x_t rows are double-buffered and fetched with async-to-LDS one step ahead.
// ---------------------------------------------------------------------------
__global__ __launch_bounds__(NTHREADS)
void rnn_fused_kernel(const int* __restrict__ q, const int* __restrict__ a,
                      const float* __restrict__ emb,
                      const float* __restrict__ conv_w,
                      const float* __restrict__ conv_b,
                      float* __restrict__ qa /* (B, 2*D*D) scratch */)
{
    __shared__ float h1[DSZ * STRIDE];
    __shared__ float h2[DSZ * STRIDE];
    __shared__ float xv[2][DSZ];
    __shared__ float uu[DSZ / 2];
    __shared__ float red[DSZ];

    const int bid    = blockIdx.x;
    const int tensor = bid / (BSZ * COLG);
    const int rem    = bid % (BSZ * COLG);
    const int b      = rem / COLG;
    const int cg     = rem % COLG;

    const int tid   = threadIdx.x;
    const int jl    = tid & (COLS - 1);        // local column 0..15
    const int rbase = (tid >> 4) * RPT;        // first owned row
    const int jg    = cg * COLS + jl;          // global column 0..127

    const int* idxp = (tensor == 0) ? q : a;

    const float w01 = conv_w[0], w11 = conv_w[1];   // layer 1
    const float w02 = conv_w[2], w12 = conv_w[3];   // layer 2
    const float b1  = conv_b[0], b2  = conv_b[1];

    for (int e = tid; e < DSZ * STRIDE; e += NTHREADS) { h1[e] = 0.0f; h2[e] = 0.0f; }
    float mx[RPT];
#pragma unroll
    for (int e = 0; e < RPT; ++e) mx[e] = -__builtin_inff();
    __syncthreads();

#ifdef HAVE_ASYNC_LDS
    // Prologue: async-fetch x_0 into buffer 0 (waves 0..3 issue, uniform branch).
    if (tid < DSZ) {
        const int idx0 = idxp[b * LSZ + 0];
        async_copy_f32(&xv[0][tid], emb + (long)idx0 * DSZ + tid);
    }
#endif

    for (int t = 0; t < LSZ; ++t) {
        const int cur = t & 1;
#ifdef HAVE_ASYNC_LDS
        __builtin_amdgcn_s_wait_asynccnt(0);   // per-wave async completion
        __syncthreads();                        // make xv[cur] visible to all
        if (t + 1 < LSZ && tid < DSZ) {
            // Overlap next gather with this step's compute. The target buffer
            // was last read in step t-1, ordered before this point by the
            // loop-bottom barrier.
            const int idxn = idxp[b * LSZ + t + 1];
            async_copy_f32(&xv[cur ^ 1][tid], emb + (long)idxn * DSZ + tid);
        }
#else
        if (tid < DSZ) {
            const int idx = idxp[b * LSZ + t];
            xv[cur][tid] = emb[(long)idx * DSZ + tid];
            if (t + 1 < LSZ) {
                const int idxn = idxp[b * LSZ + t + 1];
                __builtin_prefetch(&emb[(long)idxn * DSZ + tid], 0, 1);
            }
        }
        __syncthreads();
#endif
        const float* xq = xv[cur];

        // --- ||x||^2 tree reduction ---
        if (tid < 64) red[tid] = xq[tid] * xq[tid] + xq[tid + 64] * xq[tid + 64];
        __syncthreads();
        if (tid < 32) red[tid] += red[tid + 32];
        __syncthreads();
        if (tid < 16) red[tid] += red[tid + 16];
        __syncthreads();
        if (tid < 8)  red[tid] += red[tid + 8];
        __syncthreads();
        if (tid < 4)  red[tid] += red[tid + 4];
        __syncthreads();
        if (tid < 2)  red[tid] += red[tid + 2];
        __syncthreads();
        if (tid == 0) red[0] = 1.0f / (red[0] + red[1] + EPSV);
        __syncthreads();

        const float rinv = red[0];
        // density folded: layer-1 top half pre-activation = u[i] * x[j] + b1
        if (tid < 64) uu[tid] = (w01 * xq[2 * tid] + w11 * xq[2 * tid + 1]) * rinv;
        __syncthreads();

        const float xj = xq[jg];

        // --- layer 1 step: compute to regs, barrier, write (single buffer) ---
        float nh[RPT];
#pragma unroll
        for (int e = 0; e < RPT; ++e) {
            const int r = rbase + e;
            float pre;
            if (r < 64) {
                pre = uu[r] * xj + b1;
            } else {
                const int k = r - 64;
                pre = w01 * h1[(2 * k) * STRIDE + jl]
                    + w11 * h1[(2 * k + 1) * STRIDE + jl] + b1;
            }
            nh[e] = TANH(pre);
        }
        __syncthreads();
#pragma unroll
        for (int e = 0; e < RPT; ++e) h1[(rbase + e) * STRIDE + jl] = nh[e];
        __syncthreads();

        // --- layer 2 step (consumes fresh h1, recurses on old h2) ---
#pragma unroll
        for (int e = 0; e < RPT; ++e) {
            const int r = rbase + e;
            float pre;
            if (r < 64) {
                pre = w02 * h1[(2 * r) * STRIDE + jl]
                    + w12 * h1[(2 * r + 1) * STRIDE + jl] + b2;
            } else {
                const int k = r - 64;
                pre = w02 * h2[(2 * k) * STRIDE + jl]
                    + w12 * h2[(2 * k + 1) * STRIDE + jl] + b2;
            }
            nh[e] = TANH(pre);
        }
        __syncthreads();
#pragma unroll
        for (int e = 0; e < RPT; ++e) {
            h2[(rbase + e) * STRIDE + jl] = nh[e];
            mx[e] = fmaxf(mx[e], nh[e]);
        }
        __syncthreads();
    }

    // qa layout: [b][tensor*D*D + i*D + j]
#pragma unroll
    for (int e = 0; e < RPT; ++e) {
        const int r = rbase + e;
        qa[(long)b * (2 * DSZ * DSZ) + tensor * (DSZ * DSZ) + r * DSZ + jg] = mx[e];
    }
}

// ---------------------------------------------------------------------------
// Head: score = qa(16 x 32768) @ lin_w^T + lin_b, then 2-class log_softmax.
// True matmul shape -> V_WMMA_F32_16X16X4_F32. K split across 8 waves, two
// accumulators per wave to break the D->C WMMA dependency chain. N padded
// 2->16 with branchless masking so EXEC is all-1s at every WMMA.
// ---------------------------------------------------------------------------
__global__ __launch_bounds__(256)
void head_wmma_kernel(const float* __restrict__ qa,
                      const float* __restrict__ lin_w,
                      const float* __restrict__ lin_b,
                      float* __restrict__ out)
{
    __shared__ float partial[8 * 16 * 2];
    __shared__ float score[16 * 2];

    const int tid  = threadIdx.x;
    const int wave = tid >> 5;
    const int lane = tid & 31;
    const int m    = lane & 15;     // A: row M (=batch); B: col N
    const int hi   = lane >> 4;     // selects K pair within the 16x4 tile

    const int K    = 2 * DSZ * DSZ;     // 32768
    const int kper = K / 8;             // per-wave K slice
    const int k0   = wave * kper;

    const int   wrow  = (m < 2) ? m : 0;        // safe lin_w row
    const float wmask = (m < 2) ? 1.0f : 0.0f;  // zero the N-padding

    v8f c0 = {0.f, 0.f, 0.f, 0.f, 0.f, 0.f, 0.f, 0.f};
    v8f c1 = {0.f, 0.f, 0.f, 0.f, 0.f, 0.f, 0.f, 0.f};

    for (int kk = k0; kk < k0 + kper; kk += 8) {
        v2f av, bv;
        const int ka = kk + 2 * hi;
        av.x = qa[(long)m * K + ka];
        av.y = qa[(long)m * K + ka + 1];
        bv.x = lin_w[(long)wrow * K + ka] * wmask;
        bv.y = lin_w[(long)wrow * K + ka + 1] * wmask;
        c0 = __builtin_amdgcn_wmma_f32_16x16x4_f32(false, av, false, bv,
                                                   (short)0, c0, false, false);
        const int kb = kk + 4 + 2 * hi;
        av.x = qa[(long)m * K + kb];
        av.y = qa[(long)m * K + kb + 1];
        bv.x = lin_w[(long)wrow * K + kb] * wmask;
        bv.y = lin_w[(long)wrow * K + kb + 1] * wmask;
        c1 = __builtin_amdgcn_wmma_f32_16x16x4_f32(false, av, false, bv,
                                                   (short)0, c1, false, false);
    }
#pragma unroll
    for (int r = 0; r < 8; ++r) c0[r] += c1[r];

    // D layout: lane l, vgpr r holds (M = r + 8*(l>>4), N = l&15)
    if (m < 2) {
#pragma unroll
        for (int r = 0; r < 8; ++r) {
            const int M = r + 8 * hi;
            partial[(wave * 16 + M) * 2 + m] = c0[r];
        }
    }
    __syncthreads();

    if (tid < 32) {
        const int bb = tid >> 1, n = tid & 1;
        float s = lin_b[n];
        for (int w = 0; w < 8; ++w) s += partial[(w * 16 + bb) * 2 + n];
        score[bb * 2 + n] = s;
    }
    __syncthreads();

    if (tid < 16) {
        const float s0 = score[tid * 2], s1 = score[tid * 2 + 1];
        const float mmax = fmaxf(s0, s1);
        const float l = mmax + logf(expf(s0 - mmax) + expf(s1 - mmax));
        out[tid * 2]     = s0 - l;
        out[tid * 2 + 1] = s1 - l;
    }
}

extern "C" void kernel_launch(void* const* d_in, const int* in_sizes, int n_in,
                              void* d_out, int out_size, void* d_ws, size_t ws_size,
                              hipStream_t stream) {
    const int*   q      = (const int*)d_in[0];
    const int*   a      = (const int*)d_in[1];
    const float* emb    = (const float*)d_in[2];
    const float* conv_w = (const float*)d_in[3];
    const float* conv_b = (const float*)d_in[4];
    const float* lin_w  = (const float*)d_in[5];
    const float* lin_b  = (const float*)d_in[6];
    float*       out    = (float*)d_out;
    float*       qa     = (float*)d_ws;   // 16 * 32768 floats = 2 MB

    dim3 grid(NT * BSZ * COLG);           // 256 workgroups
    rnn_fused_kernel<<<grid, NTHREADS, 0, stream>>>(q, a, emb, conv_w, conv_b, qa);
    head_wmma_kernel<<<1, 256, 0, stream>>>(qa, lin_w, lin_b, out);
}